// LSTM_45672682226515
// MI455X (gfx1250) — compile-verified
//
#include <hip/hip_runtime.h>
#include <math.h>

// Problem shape (fixed by the reference)
#define T_STEPS 16384
#define IN_DIM  256
#define HID     512
#define GATES   2048   // 4*HID, gate order i,f,g,o

// Scan partitioning: 16 persistent workgroups, each owns 32 hidden units
// => 128 gate-rows of w_hh (4 gates x 32) resident in LDS: 2*128*260 dwords.
#define NWG   16
#define SLICE 32
#define ROWS  128
#define RSTR  260      // 256 data dwords + 4 pad dwords per half-row (bank tuning)

typedef float v2f __attribute__((ext_vector_type(2)));
typedef float v8f __attribute__((ext_vector_type(8)));

__device__ __forceinline__ float sigmoid_f(float x) {
  return 1.0f / (1.0f + __expf(-x));
}
__device__ __forceinline__ float tanh_f(float x) {
#if defined(__has_builtin) && __has_builtin(__builtin_amdgcn_tanhf)
  return __builtin_amdgcn_tanhf(x);   // gfx1250 v_tanh_f32
#else
  return tanhf(x);
#endif
}

// ---------------------------------------------------------------------------
// Phase 1: x_proj[T,2048] = inputs[T,256] @ w_ih^T + bias, fp32 WMMA 16x16x4.
// Block = 256 threads = 8 waves; each wave computes one 16x16 C tile, all 8
// waves share one 16-wide column panel of B staged in LDS.
// ---------------------------------------------------------------------------
__global__ void __launch_bounds__(256)
lstm_xproj_gemm(const float* __restrict__ inputs,
                const float* __restrict__ w_ih,   // [2048,256] row-major
                const float* __restrict__ bias,   // [2048]
                float* __restrict__ xproj)        // [T,2048]
{
  __shared__ float Bl[256][17];                   // B[k][n], padded stride 17
  const int tid   = threadIdx.x;
  const int lane  = tid & 31;
  const int wave  = tid >> 5;
  const int n0    = blockIdx.x * 16;              // output-column tile base
  const int rows0 = blockIdx.y * 128;             // 8 row-tiles per block

  // Stage B panel: B[k][n] = w_ih[(n0+n)*256 + k]; coalesced global reads.
  for (int i = tid; i < 16 * 256; i += 256) {
    const int n = i >> 8, k = i & 255;
    Bl[k][n] = w_ih[(n0 + n) * IN_DIM + k];
  }
  __syncthreads();

  const int h  = lane >> 4;                       // half-wave select
  const int nl = lane & 15;
  const int rowA = rows0 + wave * 16 + nl;        // A row fed by this lane
  const float2* Arow = (const float2*)(inputs + (size_t)rowA * IN_DIM);

  v8f acc = {};
#if defined(__has_builtin) && __has_builtin(__builtin_amdgcn_wmma_f32_16x16x4_f32)
  #pragma unroll 4
  for (int k = 0; k < IN_DIM; k += 4) {
    // A 16x4 layout: VGPR0 = K (k+2h), VGPR1 = K (k+2h+1)
    const float2 a = Arow[(k >> 1) + h];
    const v2f av = {a.x, a.y};
    // B 4x16 layout: VGPR0 = rows k (lanes 0-15) / k+2 (lanes 16-31), VGPR1 = k+1 / k+3
    const v2f bv = {Bl[k + 2 * h][nl], Bl[k + 1 + 2 * h][nl]};
    acc = __builtin_amdgcn_wmma_f32_16x16x4_f32(
        /*neg_a=*/false, av, /*neg_b=*/false, bv,
        /*c_mod=*/(short)0, acc, /*reuse_a=*/false, /*reuse_b=*/false);
  }
#else
  // Scalar fallback (should not be taken on gfx1250 toolchains).
  #pragma unroll 4
  for (int k = 0; k < IN_DIM; k += 4) {
    const float2 a = Arow[(k >> 1) + h];
    acc[0] = fmaf(a.x, Bl[k + 2 * h][nl], acc[0]);
    acc[1] = fmaf(a.y, Bl[k + 1 + 2 * h][nl], acc[1]);
  }
#endif

  const float bb = bias[n0 + nl];
  float* outBase = xproj + (size_t)(rows0 + wave * 16) * GATES + n0 + nl;
  // C/D layout: VGPR r holds rows r (lanes 0-15) and r+8 (lanes 16-31)
  #pragma unroll
  for (int r = 0; r < 8; ++r) {
    outBase[(size_t)(r + 8 * h) * GATES] = acc[r] + bb;
  }
}

// ---------------------------------------------------------------------------
// Phase 2: persistent recurrent scan. 16 WGs x 256 threads; each WG keeps its
// 128 w_hh rows (256 KB, padded) in LDS for all 16384 steps; h is exchanged
// through L2 each step behind a cluster-barrier + atomic grid barrier.
// ---------------------------------------------------------------------------
__global__ void __launch_bounds__(256)
lstm_scan(const float* __restrict__ w_hh,   // [2048,512]
          const float* __restrict__ xproj,  // [T,2048]
          float* __restrict__ c_out,        // [T,512]
          float* __restrict__ hbuf,         // [2,512] double-buffered h
          unsigned* __restrict__ bar)       // grid barrier counter (zeroed)
{
  extern __shared__ float smem[];
  float* Wl   = smem;                        // [2][ROWS][RSTR]
  float* hl   = smem + 2 * ROWS * RSTR;      // [HID]
  float* psum = hl + HID;                    // [256]
  float* actl = psum + 256;                  // [128]

  const int tid = threadIdx.x;
  const int j0  = blockIdx.x * SLICE;        // hidden-unit slice base

  // ---- Preload w_hh slice into padded LDS layout Wl[half][r][0..255] ------
  // local row r -> global gate-row (r>>5)*HID + j0 + (r&31)
#if defined(__has_builtin) && __has_builtin(__builtin_amdgcn_tensor_load_to_lds)
  if (tid < 32) {   // wave 0 drives the Tensor Data Mover (8 descriptors)
    typedef unsigned u32x4 __attribute__((ext_vector_type(4)));
    typedef int      i32x4 __attribute__((ext_vector_type(4)));
    typedef int      i32x8 __attribute__((ext_vector_type(8)));
    #pragma unroll
    for (int g = 0; g < 4; ++g) {
      #pragma unroll
      for (int hh = 0; hh < 2; ++hh) {
        const unsigned ldsOff =
            (unsigned)(uintptr_t)(Wl + ((size_t)hh * ROWS + g * SLICE) * RSTR);
        const unsigned long long ga = (unsigned long long)(uintptr_t)
            (w_hh + ((size_t)g * HID + j0) * HID + hh * 256);
        // D# group0: count=1, lds_addr, 57-bit global addr, type=2
        const u32x4 g0 = { 1u, ldsOff,
                           (unsigned)(ga & 0xffffffffu),
                           (unsigned)((ga >> 32) & 0x01ffffffu) | 0x80000000u };
        // D# group1: data_size=4B, pad_enable, pad every 256 dw by 4 dw,
        // tensor_dim0=256, tensor_dim1=32, tile 256x32, dim0 stride=512
        const i32x8 g1 = { (int)((2u << 16) | (1u << 20) | (7u << 22) | (3u << 25)),
                           (int)(256u << 16),   // tensor_dim0[15:0]
                           (int)(32u  << 16),   // tensor_dim1[15:0]
                           (int)(256u << 16),   // tile_dim0
                           32,                  // tile_dim1
                           512,                 // tensor_dim0_stride lo
                           0, 0 };
        const i32x4 z4 = {0, 0, 0, 0};
#if __clang_major__ >= 23
        const i32x8 z8 = {0, 0, 0, 0, 0, 0, 0, 0};
        __builtin_amdgcn_tensor_load_to_lds(g0, g1, z4, z4, z8, 0);
#else
        __builtin_amdgcn_tensor_load_to_lds(g0, g1, z4, z4, 0);
#endif
      }
    }
#if defined(__has_builtin) && __has_builtin(__builtin_amdgcn_s_wait_tensorcnt)
    __builtin_amdgcn_s_wait_tensorcnt(0);
#else
    asm volatile("s_wait_tensorcnt 0x0" ::: "memory");
#endif
  }
#else
  for (int i = tid; i < ROWS * HID; i += 256) {
    const int r = i >> 9, k = i & 511;
    const int grow = (r >> 5) * HID + j0 + (r & 31);
    Wl[((size_t)(k >> 8) * ROWS + r) * RSTR + (k & 255)] = w_hh[(size_t)grow * HID + k];
  }
#endif

  for (int i = tid; i < HID; i += 256) hl[i] = 0.0f;   // h_{-1} = 0
  __syncthreads();

  const int half = tid >> 7;                 // which 256-wide K half
  const int r    = tid & 127;                // local gate-row
  const float4* wrow = (const float4*)(Wl + ((size_t)half * ROWS + r) * RSTR);
  const float4* hvec = (const float4*)(hl + half * 256);
  const int xrow = (r >> 5) * HID + j0 + (r & 31);   // column in x_proj

  float c_reg = 0.0f;                        // cell state (valid tid < 32)

  for (int t = 0; t < T_STEPS; ++t) {
    // Prefetch this step's input projection (hides under the dot product).
    float xp = 0.0f;
    if (tid < 128) xp = xproj[(size_t)t * GATES + xrow];

    // 256-element fp32 dot product per lane, float4 LDS reads.
    float4 s = make_float4(0.f, 0.f, 0.f, 0.f);
    #pragma unroll 8
    for (int i = 0; i < 64; ++i) {
      const float4 a = wrow[i];
      const float4 b = hvec[i];
      s.x = fmaf(a.x, b.x, s.x);
      s.y = fmaf(a.y, b.y, s.y);
      s.z = fmaf(a.z, b.z, s.z);
      s.w = fmaf(a.w, b.w, s.w);
    }
    psum[tid] = (s.x + s.y) + (s.z + s.w);
    __syncthreads();

    if (tid < 128) {
      const float pre = psum[tid] + psum[tid + 128] + xp;
      actl[tid] = ((tid >> 5) == 2) ? tanh_f(pre) : sigmoid_f(pre);  // g vs i,f,o
    }
    __syncthreads();

    if (tid < SLICE) {
      const float iv = actl[tid];
      const float fv = actl[32 + tid];
      const float gv = actl[64 + tid];
      const float ov = actl[96 + tid];
      const float c  = fmaf(fv, c_reg, iv * gv);
      c_reg = c;
      c_out[(size_t)t * HID + j0 + tid] = c;               // c_seq output
      hbuf[(size_t)(t & 1) * HID + j0 + tid] = ov * tanh_f(c);
    }

    // Publish h slice, then synchronize the 16-WG grid for this step.
    __threadfence();
#if defined(__has_builtin) && __has_builtin(__builtin_amdgcn_s_cluster_barrier)
    __builtin_amdgcn_s_cluster_barrier();   // NOP unless cluster-launched
#endif
    if (tid == 0) {
      __hip_atomic_fetch_add(bar, 1u, __ATOMIC_RELEASE, __HIP_MEMORY_SCOPE_AGENT);
      const unsigned target = (unsigned)(t + 1) * NWG;
      while (__hip_atomic_load(bar, __ATOMIC_ACQUIRE, __HIP_MEMORY_SCOPE_AGENT) < target) {
        __builtin_amdgcn_s_sleep(2);
      }
    }
    __syncthreads();

    // Pull the full broadcast h_t for the next step.
    {
      const float* hsrc = hbuf + (size_t)(t & 1) * HID;
      for (int i = tid; i < HID; i += 256) hl[i] = hsrc[i];
    }
    __syncthreads();
  }
}

// ---------------------------------------------------------------------------
// Workspace layout (assumes ws_size >= ~129 MB):
//   [0, 128MB)        x_proj [16384, 2048] f32
//   +128MB, 4KB       h double buffer [2, 512] f32
//   +128MB+4KB, 64B   grid-barrier counter
// ---------------------------------------------------------------------------
extern "C" void kernel_launch(void* const* d_in, const int* in_sizes, int n_in,
                              void* d_out, int out_size, void* d_ws, size_t ws_size,
                              hipStream_t stream) {
  const float* inputs = (const float*)d_in[0];
  const float* w_ih   = (const float*)d_in[1];
  const float* w_hh   = (const float*)d_in[2];
  const float* bias   = (const float*)d_in[3];
  float* c_out        = (float*)d_out;

  char* ws            = (char*)d_ws;
  const size_t xp_b   = (size_t)T_STEPS * GATES * sizeof(float);
  float* xproj        = (float*)ws;
  float* hbuf         = (float*)(ws + xp_b);
  unsigned* bar       = (unsigned*)(ws + xp_b + 4096);

  // Barrier counter must start at 0 every launch (d_ws is poisoned once,
  // never restored) — capture-safe async memset.
  hipMemsetAsync(bar, 0, 64, stream);

  dim3 grid1(GATES / 16, T_STEPS / 128);
  lstm_xproj_gemm<<<grid1, 256, 0, stream>>>(inputs, w_ih, bias, xproj);

  const size_t lds_bytes =
      (size_t)(2 * ROWS * RSTR + HID + 256 + 128) * sizeof(float); // 269,824 B
  lstm_scan<<<NWG, 256, lds_bytes, stream>>>(w_hh, xproj, c_out, hbuf, bar);
}